// _ESHolt_2748779069988
// MI455X (gfx1250) — compile-verified
//
#include <hip/hip_runtime.h>

// ES-Holt (Holt-Winters) scan: B=32768 series, T=512 steps, S=24 seasons.
// Memory-bound (~270MB traffic -> ~12us floor @ 23.3TB/s). One lane per
// series; season ring buffer kept in VGPRs via static 24-step unrolling
// (verified: no scratch ops, buffer lives in v[2:25]); all global I/O is
// float4/b128; v_rcp_f32/v_exp_f32 fast math; global_prefetch_b8 for the y
// stream; non-temporal stores keep the 192MB L2 free for the y read stream.

#define BB 32768
#define TT 512
#define SS 24
#define TSL 536  // TT + SS

typedef float v4f __attribute__((ext_vector_type(4)));  // native vector for NT builtin

__device__ __forceinline__ float fast_rcp(float x) {
  return __builtin_amdgcn_rcpf(x);            // v_rcp_f32
}
__device__ __forceinline__ float fast_exp(float x) {
  return __builtin_amdgcn_exp2f(x * 1.44269504088896340736f);  // v_exp_f32
}
__device__ __forceinline__ float fast_sigmoid(float x) {
  return fast_rcp(1.0f + fast_exp(-x));
}

// Non-temporal 16B store: outputs are write-once streams, never re-read.
__device__ __forceinline__ void nt_store4(float* p, float a, float b, float c,
                                          float d) {
  v4f v = {a, b, c, d};
  __builtin_nontemporal_store(v, (v4f*)p);  // global_store_b128 ... nt
}

// One recurrence step. tcur is dynamic; POS (t%24) and Q (t&3) must be
// compile-time constants (supplied from fully-unrolled loop indices) so the
// season buffer stays in registers and the float4 lane select folds.
#define STEP(tcur, POS, Q)                                                    \
  {                                                                           \
    if ((Q) == 0) { yv = *(const float4*)(yb + (tcur)); }                     \
    const float y_t = (Q) == 0 ? yv.x : (Q) == 1 ? yv.y                       \
                      : (Q) == 2 ? yv.z : yv.w;                               \
    const float seas = buf[(POS)];                                            \
    const float newlev = al * (y_t * fast_rcp(seas)) + bl * (lev + tr);       \
    const float newtr  = as_ * (newlev - lev) + bs * tr; /* a_trend==a_seas */\
    const float newseas = as_ * (y_t * fast_rcp(newlev + newtr)) + bs * seas; \
    buf[(POS)] = newseas;                                                     \
    lev = newlev; tr = newtr;                                                 \
    Lb[(Q)] = newlev; Tb[(Q)] = newtr; Sb[(Q)] = newseas;                     \
    if ((Q) == 3) {                                                           \
      nt_store4(levrow + (tcur) - 3, Lb[0], Lb[1], Lb[2], Lb[3]);             \
      nt_store4(trrow + (tcur) - 3, Tb[0], Tb[1], Tb[2], Tb[3]);              \
      nt_store4(searow + 21 + (tcur), Sb[0], Sb[1], Sb[2], Sb[3]);            \
    }                                                                         \
  }

__global__ __launch_bounds__(128) void es_holt_kernel(
    const float* __restrict__ y,          // [B, T]
    const int* __restrict__ idxs,         // [B]
    const float* __restrict__ lev_sms,    // [B, 1]
    const float* __restrict__ seas_sms,   // [B, 1]
    const float* __restrict__ init_seas,  // [B, S]
    float* __restrict__ levels,           // [B, T]
    float* __restrict__ seasonalities,    // [B, T+S]
    float* __restrict__ trends) {         // [B, T]
  const int b = blockIdx.x * 128 + threadIdx.x;
  if (b >= BB) return;

  const int idx = idxs[b];
  const float al = fast_sigmoid(lev_sms[idx]);
  const float as_ = fast_sigmoid(seas_sms[idx]);  // a_trend = a_seas (source bug kept)
  const float bl = 1.0f - al;
  const float bs = 1.0f - as_;

  const float* __restrict__ yb = y + (size_t)b * TT;
  float* __restrict__ levrow = levels + (size_t)b * TT;
  float* __restrict__ trrow = trends + (size_t)b * TT;
  float* __restrict__ searow = seasonalities + (size_t)b * TSL;

  // Season ring buffer in registers; also emit seasonalities[b, 0:24].
  float buf[SS];
  {
    const float4* __restrict__ ip =
        (const float4*)(init_seas + (size_t)idx * SS);  // 96B offset -> 16B aligned
#pragma unroll
    for (int i = 0; i < SS / 4; ++i) {
      const float4 v = ip[i];
      const float e0 = fast_exp(v.x), e1 = fast_exp(v.y);
      const float e2 = fast_exp(v.z), e3 = fast_exp(v.w);
      buf[4 * i + 0] = e0; buf[4 * i + 1] = e1;
      buf[4 * i + 2] = e2; buf[4 * i + 3] = e3;
      nt_store4(searow + 4 * i, e0, e1, e2, e3);
    }
  }

  // t = 0 initialization.
  float4 yv = *(const float4*)yb;  // y[b, 0:4]
  float lev = yv.x * fast_rcp(buf[0]);
  float tr = (yv.y * fast_rcp(buf[1]) - lev) * 0.5f;
  float Lb[4], Tb[4], Sb[4];
  Lb[0] = lev;     // levels[b,0]
  Tb[0] = tr;      // trends[b,0]
  Sb[0] = buf[0];  // seasonalities[b,24] = exp(init_seas)[b,0]

  // t = 1 .. 504 : 21 chunks of 24 (LCM of ring period 24 and vector phase 4).
  for (int k = 0; k < 21; ++k) {
    const int t0 = 1 + 24 * k;
    // gfx1250 prefetch path: pull the y stream ~4 chunks ahead, clamped in-row.
    const int pft = (t0 + 96 <= TT - 4) ? (t0 + 96) : (TT - 4);
    __builtin_prefetch((const void*)(yb + pft), 0, 1);  // -> global_prefetch_b8
#pragma unroll
    for (int j = 0; j < 24; ++j) {
      STEP(t0 + j, (j + 1) % 24, (j + 1) & 3);  // t%24=(j+1)%24, t&3=(j+1)&3
    }
  }
  // t = 505 .. 511 tail (pos = 1..7, phase continues; ends on q==3 at t=511).
#pragma unroll
  for (int j = 0; j < 7; ++j) {
    STEP(505 + j, j + 1, (j + 1) & 3);
  }
}

extern "C" void kernel_launch(void* const* d_in, const int* in_sizes, int n_in,
                              void* d_out, int out_size, void* d_ws, size_t ws_size,
                              hipStream_t stream) {
  (void)in_sizes; (void)n_in; (void)d_ws; (void)ws_size; (void)out_size;
  const float* y = (const float*)d_in[0];
  const int* idxs = (const int*)d_in[1];
  const float* lev_sms = (const float*)d_in[2];
  const float* seas_sms = (const float*)d_in[3];
  // d_in[4] (trend_sms) intentionally unused: reference replicates the source
  // bug where a_trend is taken from seas_sms.
  const float* init_seas = (const float*)d_in[5];

  float* levels = (float*)d_out;                       // [B, T]
  float* seasonal = levels + (size_t)BB * TT;          // [B, T+S]
  float* trends = seasonal + (size_t)BB * TSL;         // [B, T]

  dim3 grid(BB / 128), block(128);
  hipLaunchKernelGGL(es_holt_kernel, grid, block, 0, stream,
                     y, idxs, lev_sms, seas_sms, init_seas,
                     levels, seasonal, trends);
}